// Neuron_9388798509385
// MI455X (gfx1250) — compile-verified
//
#include <hip/hip_runtime.h>
#include <stdint.h>

// Problem constants (fixed by the reference): x/noise/unif are [B, T, N] f32.
namespace {
constexpr int B     = 32;
constexpr int T     = 16;
constexpr int N     = 65536;
constexpr int TPB   = 256;          // 8 wave32 per block
constexpr int VEC   = 4;            // floats per thread (b128 transactions)
constexpr int TILE  = TPB * VEC;    // 1024 floats of N per block per timestep
constexpr int DEPTH = 4;            // async prefetch pipeline depth (timesteps)
}

// Stochastic LIF scan, one thread owns 4 consecutive n-lanes, u kept in VGPRs.
// Data for timestep t is staged LDS via CDNA5 async copies, DEPTH timesteps deep.
__global__ __launch_bounds__(TPB) void snn_stoch_lif_kernel(
    const float* __restrict__ x,
    const float* __restrict__ noise,
    const float* __restrict__ unif,
    float* __restrict__ out) {
  // [depth][array: x/noise/unif][thread] staging buffers, 48 KB total.
  __shared__ float4 smem[DEPTH][3][TPB];

  const int tid = threadIdx.x;
  const int b   = blockIdx.y;
  const int n0  = blockIdx.x * TILE + tid * VEC;

  // Low 32 bits of a flat shared-aperture address == LDS byte offset.
  const uint32_t lds_base = (uint32_t)(uintptr_t)&smem[0][0][0];
  // Per-lane byte offset of this thread's 16B chunk inside one (b,t) slab.
  const uint32_t voff_n = (uint32_t)n0 * 4u;

  // Issue the 3 async tile loads for timestep `s` into buffer s % DEPTH.
  // s_wait_dscnt 0 guards the WAR hazard of refilling a buffer whose ds_load
  // readback may still be in flight. Total bytes offsets fit in u32 (128 MiB).
#define ISSUE_STAGE(s)                                                         \
  {                                                                            \
    const int      d_    = (s) & (DEPTH - 1);                                  \
    const uint32_t goff_ = ((uint32_t)(b * T + (s)) << 18) + voff_n;           \
    const uint32_t l0_ = lds_base + (uint32_t)(((d_ * 3 + 0) * TPB + tid) * 16);\
    const uint32_t l1_ = lds_base + (uint32_t)(((d_ * 3 + 1) * TPB + tid) * 16);\
    const uint32_t l2_ = lds_base + (uint32_t)(((d_ * 3 + 2) * TPB + tid) * 16);\
    asm volatile(                                                              \
        "s_wait_dscnt 0\n\t"                                                   \
        "global_load_async_to_lds_b128 %0, %3, %4\n\t"                         \
        "global_load_async_to_lds_b128 %1, %3, %5\n\t"                         \
        "global_load_async_to_lds_b128 %2, %3, %6"                             \
        :                                                                      \
        : "v"(l0_), "v"(l1_), "v"(l2_), "v"(goff_),                            \
          "s"(x), "s"(noise), "s"(unif)                                        \
        : "memory");                                                           \
  }

  // Prologue: fill the pipeline (T >= DEPTH always here).
  ISSUE_STAGE(0);
  ISSUE_STAGE(1);
  ISSUE_STAGE(2);
  ISSUE_STAGE(3);

  float4 u = make_float4(0.0f, 0.0f, 0.0f, 0.0f);

  for (int t = 0; t < T; ++t) {
    // Outstanding async loads at loop top: 3 * min(DEPTH, T - t).
    // Async loads complete in order, so waiting to <= 3*(stages beyond t)
    // guarantees stage t has fully landed in LDS. Exact tail counts matter.
    const int remaining = T - 1 - t;
    if (remaining >= DEPTH - 1)
      asm volatile("s_wait_asynccnt 9" ::: "memory");
    else if (remaining == 2)
      asm volatile("s_wait_asynccnt 6" ::: "memory");
    else if (remaining == 1)
      asm volatile("s_wait_asynccnt 3" ::: "memory");
    else
      asm volatile("s_wait_asynccnt 0" ::: "memory");

    const int d = t & (DEPTH - 1);
    const float4 xv = smem[d][0][tid];
    const float4 nv = smem[d][1][tid];
    const float4 rv = smem[d][2][tid];

    // u = 0.5*u + x - noise;  q = ((u-1)+0.4)/0.8;  spike = (r < q)
    // (clip(q,0,1) is redundant vs r in [0,1));  u = spike ? 0 : u.
    float4 ov;
    {
      float uu = fmaf(0.5f, u.x, xv.x) - nv.x;
      bool  sp = rv.x < ((uu - 1.0f) + 0.4f) / 0.8f;
      ov.x = sp ? 1.0f : 0.0f;
      u.x  = sp ? 0.0f : uu;
    }
    {
      float uu = fmaf(0.5f, u.y, xv.y) - nv.y;
      bool  sp = rv.y < ((uu - 1.0f) + 0.4f) / 0.8f;
      ov.y = sp ? 1.0f : 0.0f;
      u.y  = sp ? 0.0f : uu;
    }
    {
      float uu = fmaf(0.5f, u.z, xv.z) - nv.z;
      bool  sp = rv.z < ((uu - 1.0f) + 0.4f) / 0.8f;
      ov.z = sp ? 1.0f : 0.0f;
      u.z  = sp ? 0.0f : uu;
    }
    {
      float uu = fmaf(0.5f, u.w, xv.w) - nv.w;
      bool  sp = rv.w < ((uu - 1.0f) + 0.4f) / 0.8f;
      ov.w = sp ? 1.0f : 0.0f;
      u.w  = sp ? 0.0f : uu;
    }

    // Coalesced 128-bit store of the spike train.
    reinterpret_cast<float4*>(out)[(size_t)(b * T + t) * (N / 4) + (n0 >> 2)] = ov;

    // Steady state: refill the buffer we just consumed with stage t+DEPTH.
    if (t + DEPTH < T) ISSUE_STAGE(t + DEPTH);
  }
#undef ISSUE_STAGE
}

extern "C" void kernel_launch(void* const* d_in, const int* in_sizes, int n_in,
                              void* d_out, int out_size, void* d_ws, size_t ws_size,
                              hipStream_t stream) {
  (void)in_sizes; (void)n_in; (void)d_ws; (void)ws_size; (void)out_size;
  const float* x     = (const float*)d_in[0];
  const float* noise = (const float*)d_in[1];
  const float* unif  = (const float*)d_in[2];
  float* out         = (float*)d_out;

  dim3 grid(N / TILE, B);   // (64, 32) -> 2048 blocks, 512K threads
  dim3 block(TPB);
  snn_stoch_lif_kernel<<<grid, block, 0, stream>>>(x, noise, unif, out);
}